// GIN_84344567759038
// MI455X (gfx1250) — compile-verified
//
#include <hip/hip_runtime.h>
#include <hip/hip_bf16.h>

typedef __attribute__((ext_vector_type(2))) float v2f;
typedef __attribute__((ext_vector_type(8))) float v8f;

// ---------------------------------------------------------------------------
// s[i] = (1 + eps) * x[i]   (vectorized float4 over N*F elements)
// ---------------------------------------------------------------------------
__global__ void gin_init_agg(float* __restrict__ s, const float* __restrict__ x,
                             const float* __restrict__ epsp, long long n4) {
    long long i = (long long)blockIdx.x * blockDim.x + threadIdx.x;
    if (i >= n4) return;
    const float sc = 1.0f + epsp[0];
    float4 v = reinterpret_cast<const float4*>(x)[i];
    v.x *= sc; v.y *= sc; v.z *= sc; v.w *= sc;
    reinterpret_cast<float4*>(s)[i] = v;
}

// ---------------------------------------------------------------------------
// scatter: s[dst] += x[src] for every edge; one thread = one float4 feature
// chunk of one edge -> coalesced gather + global_atomic_add_f32 scatter.
// ---------------------------------------------------------------------------
__global__ void gin_scatter_add(float* __restrict__ s, const float* __restrict__ x,
                                const long long* __restrict__ ei,
                                int n_edges, int fdiv4) {
    long long t = (long long)blockIdx.x * blockDim.x + threadIdx.x;
    long long total = (long long)n_edges * fdiv4;
    if (t >= total) return;
    const int e = (int)(t / fdiv4);
    const int c = (int)(t % fdiv4);
    const long long src = ei[e];
    const long long dst = ei[(long long)n_edges + e];
    const int F = fdiv4 * 4;
    const float4 v = reinterpret_cast<const float4*>(x + src * F)[c];
    float* p = s + dst * F + (long long)c * 4;
    atomicAdd(p + 0, v.x);
    atomicAdd(p + 1, v.y);
    atomicAdd(p + 2, v.z);
    atomicAdd(p + 3, v.w);
}

// ---------------------------------------------------------------------------
// C[N x 128] = act(A[N x K] @ W[K x 128] + bias), act = ReLU, then optional
// eval-mode BatchNorm folded to per-column scale/shift.
// One wave -> one 16x16 tile via V_WMMA_F32_16X16X4_F32; 8 waves/block cover
// all 128 output columns of a 16-row slab. Requires N % 16 == 0 (full EXEC).
// ---------------------------------------------------------------------------
__global__ void gin_gemm_wmma(const float* __restrict__ A,
                              const float* __restrict__ W,
                              const float* __restrict__ bias,
                              const float* __restrict__ gamma,
                              const float* __restrict__ beta,
                              const float* __restrict__ mean,
                              const float* __restrict__ var,
                              float* __restrict__ C, int K) {
    const int lane = threadIdx.x & 31;
    const int wave = threadIdx.x >> 5;      // 0..7 : output column tile
    const int row0 = blockIdx.x * 16;
    const int col0 = wave * 16;
    const int half = lane >> 4;             // 0 or 1
    const int l15  = lane & 15;

    // A fragment layout (ISA 7.12.2, 32-bit A 16x4): lane holds row M=lane%16;
    // VGPR0/VGPR1 = K{0,1} (lanes 0-15) or K{2,3} (lanes 16-31).
    const float2* A2 = reinterpret_cast<const float2*>(A + (size_t)(row0 + l15) * K);
    // B fragment: row K striped across lanes within a VGPR (col = lane%16).
    const float* Bcol = W + col0 + l15;

    v8f acc = {};
    for (int k0 = 0; k0 < K; k0 += 4) {
        const float2 af = A2[(k0 >> 1) + half];
        v2f a; a[0] = af.x; a[1] = af.y;
        const int kb = k0 + 2 * half;
        v2f b; b[0] = Bcol[(size_t)kb * 128]; b[1] = Bcol[(size_t)(kb + 1) * 128];
        // (neg_a, A, neg_b, B, c_mod, C, reuse_a, reuse_b)
        acc = __builtin_amdgcn_wmma_f32_16x16x4_f32(false, a, false, b,
                                                    (short)0, acc, false, false);
    }

    const int col = col0 + l15;
    const float fb = bias[col];
    float scale = 1.0f, shift = 0.0f;
    if (gamma != nullptr) {
        scale = gamma[col] * rsqrtf(var[col] + 1e-5f);
        shift = beta[col] - mean[col] * scale;
    }
    #pragma unroll
    for (int r = 0; r < 8; ++r) {
        const int row = row0 + r + 8 * half;   // C/D layout: VGPR r, lane half
        float v = acc[r] + fb;
        v = v > 0.0f ? v : 0.0f;               // ReLU
        if (gamma != nullptr) v = v * scale + shift;  // BN after ReLU (ref order)
        C[(size_t)row * 128 + col] = v;
    }
}

// ---------------------------------------------------------------------------
// out[n] = dot(h[n, 0:128], Wf) + bf ; one wave per node, float4 per lane,
// wave32 xor-shuffle reduction.
// ---------------------------------------------------------------------------
__global__ void gin_head_dot(const float* __restrict__ h,
                             const float* __restrict__ Wf,
                             const float* __restrict__ bf,
                             float* __restrict__ out, int N) {
    const int lane = threadIdx.x & 31;
    const int wave = threadIdx.x >> 5;
    const int node = blockIdx.x * 8 + wave;
    if (node >= N) return;
    const float4 hv = reinterpret_cast<const float4*>(h + (size_t)node * 128)[lane];
    const float4 wv = reinterpret_cast<const float4*>(Wf)[lane];
    float p = hv.x * wv.x + hv.y * wv.y + hv.z * wv.z + hv.w * wv.w;
    #pragma unroll
    for (int off = 16; off > 0; off >>= 1) p += __shfl_xor(p, off, 32);
    if (lane == 0) out[node] = p + bf[0];
}

// ---------------------------------------------------------------------------
extern "C" void kernel_launch(void* const* d_in, const int* in_sizes, int n_in,
                              void* d_out, int out_size, void* d_ws, size_t ws_size,
                              hipStream_t stream) {
    const float*      x    = (const float*)d_in[0];
    const long long*  ei   = (const long long*)d_in[1];   // int64 edge_index [2,E]
    const float* eps1 = (const float*)d_in[2];
    const float* W1a  = (const float*)d_in[3];
    const float* b1a  = (const float*)d_in[4];
    const float* W1b  = (const float*)d_in[5];
    const float* b1b  = (const float*)d_in[6];
    const float* g1   = (const float*)d_in[7];
    const float* be1  = (const float*)d_in[8];
    const float* m1   = (const float*)d_in[9];
    const float* v1   = (const float*)d_in[10];
    const float* eps2 = (const float*)d_in[11];
    const float* W2a  = (const float*)d_in[12];
    const float* b2a  = (const float*)d_in[13];
    const float* g2   = (const float*)d_in[14];
    const float* be2  = (const float*)d_in[15];
    const float* m2   = (const float*)d_in[16];
    const float* v2   = (const float*)d_in[17];
    const float* eps3 = (const float*)d_in[18];
    const float* W3a  = (const float*)d_in[19];
    const float* b3a  = (const float*)d_in[20];
    const float* W3b  = (const float*)d_in[21];
    const float* b3b  = (const float*)d_in[22];
    const float* g3   = (const float*)d_in[23];
    const float* be3  = (const float*)d_in[24];
    const float* m3   = (const float*)d_in[25];
    const float* v3   = (const float*)d_in[26];
    const float* Wl   = (const float*)d_in[27];
    const float* bl   = (const float*)d_in[28];
    const float* Wf   = (const float*)d_in[29];
    const float* bf   = (const float*)d_in[30];
    float* out = (float*)d_out;

    const int N = in_sizes[0] / 64;     // 100000 (multiple of 16)
    const int E = in_sizes[1] / 2;      // 1600000

    // Ping-pong buffers in workspace, each N x 128 f32.
    float* P = (float*)d_ws;
    float* Q = P + (size_t)N * 128;

    const int TB = 256;
    const int gemmGrid = N / 16;                       // full 16-row slabs
    auto cdiv = [](long long a, long long b) { return (int)((a + b - 1) / b); };

    // ---- GINConv 1: agg(64) -> Lin(64,128)+ReLU -> Lin(128,128)+ReLU -> BN
    gin_init_agg<<<cdiv((long long)N * 16, TB), TB, 0, stream>>>(P, x, eps1, (long long)N * 16);
    gin_scatter_add<<<cdiv((long long)E * 16, TB), TB, 0, stream>>>(P, x, ei, E, 16);
    gin_gemm_wmma<<<gemmGrid, TB, 0, stream>>>(P, W1a, b1a, nullptr, nullptr, nullptr, nullptr, Q, 64);
    gin_gemm_wmma<<<gemmGrid, TB, 0, stream>>>(Q, W1b, b1b, g1, be1, m1, v1, P, 128);

    // ---- GINConv 2: agg(128) -> Lin+ReLU -> BN
    gin_init_agg<<<cdiv((long long)N * 32, TB), TB, 0, stream>>>(Q, P, eps2, (long long)N * 32);
    gin_scatter_add<<<cdiv((long long)E * 32, TB), TB, 0, stream>>>(Q, P, ei, E, 32);
    gin_gemm_wmma<<<gemmGrid, TB, 0, stream>>>(Q, W2a, b2a, g2, be2, m2, v2, P, 128);

    // ---- GINConv 3: agg(128) -> Lin+ReLU -> Lin+ReLU -> BN
    gin_init_agg<<<cdiv((long long)N * 32, TB), TB, 0, stream>>>(Q, P, eps3, (long long)N * 32);
    gin_scatter_add<<<cdiv((long long)E * 32, TB), TB, 0, stream>>>(Q, P, ei, E, 32);
    gin_gemm_wmma<<<gemmGrid, TB, 0, stream>>>(Q, W3a, b3a, nullptr, nullptr, nullptr, nullptr, P, 128);
    gin_gemm_wmma<<<gemmGrid, TB, 0, stream>>>(P, W3b, b3b, g3, be3, m3, v3, Q, 128);

    // ---- head: Lin+ReLU then 128->1 dot
    gin_gemm_wmma<<<gemmGrid, TB, 0, stream>>>(Q, Wl, bl, nullptr, nullptr, nullptr, nullptr, P, 128);
    gin_head_dot<<<cdiv(N, 8), TB, 0, stream>>>(P, Wf, bf, out, N);
}